// VideoIntrBonus_15324443312990
// MI455X (gfx1250) — compile-verified
//
#include <hip/hip_runtime.h>
#include <hip/hip_bf16.h>

// Problem constants (from reference)
#define FEAT_D 1536
#define QD     64       // queue_dim
#define TFULL  513
#define TT     512      // t = T - L + 1
#define NB     16       // batch
#define NTOT   8192     // NB * TT
#define KNN    16
#define CHUNK  128      // columns per chunk (8 waves x 16)
#define DPAD   20       // LDS pad: stride 148 -> conflict-free reads & writes

typedef float v2f __attribute__((ext_vector_type(2)));
typedef float v8f __attribute__((ext_vector_type(8)));

// ---------------------------------------------------------------------------
// Kernel A: win = sliding mean(L=2) of feat; sf = win @ proj ; sq = |sf_row|^2
// 256 threads = 4 rows x 64 cols
// ---------------------------------------------------------------------------
__global__ __launch_bounds__(256) void vib_proj_kernel(
    const float* __restrict__ feat, const float* __restrict__ proj,
    float* __restrict__ sf, float* __restrict__ sq) {
  __shared__ float winS[4][FEAT_D];
  __shared__ float red[4][QD];
  const int tid = threadIdx.x;
  const int r = tid >> 6;   // row within block (0..3)
  const int e = tid & 63;   // output column (0..63)
  const int n = blockIdx.x * 4 + r;
  const int b = n >> 9;     // n / 512
  const int i = n & 511;    // n % 512
  const float* f0 = feat + (size_t)(b * TFULL + i) * FEAT_D;

  for (int d = e; d < FEAT_D; d += 64)
    winS[r][d] = 0.5f * (f0[d] + f0[d + FEAT_D]);
  __syncthreads();

  float acc = 0.0f;
  #pragma unroll 8
  for (int d = 0; d < FEAT_D; ++d)
    acc = fmaf(winS[r][d], proj[d * QD + e], acc);

  sf[(size_t)n * QD + e] = acc;
  red[r][e] = acc * acc;
  __syncthreads();
  if (e == 0) {
    float s = 0.0f;
    #pragma unroll
    for (int k = 0; k < QD; ++k) s += red[r][k];
    sq[n] = s;
  }
}

// ---------------------------------------------------------------------------
// Kernel B: fused Gram(WMMA f32 16x16x4) + distance + per-row top-16 mean.
// One workgroup (256 thr = 8 waves) per 16-row tile. Waves each own a 16-col
// N-tile inside a 128-col chunk; distances staged in LDS; 16 threads/row keep
// register top-16 lists across all chunks, merged once at the end.
// ---------------------------------------------------------------------------
__global__ __launch_bounds__(256) void vib_knn_kernel(
    const float* __restrict__ sf, const float* __restrict__ sq,
    float* __restrict__ raw) {
  __shared__ float dch[16][CHUNK + DPAD];   // distance chunk, padded
  __shared__ float mergeBuf[16][16 * KNN + 1];

  const int tid  = threadIdx.x;
  const int wave = tid >> 5;          // 0..7
  const int lane = tid & 31;
  const int m    = lane & 15;         // matrix row/col index within half-wave
  const int kh   = lane >> 4;         // k half-select (A/B layout)
  const int rowBase = blockIdx.x * 16;

  // A fragments (rows rowBase..rowBase+15, K=0..63): loop-invariant, registers.
  const float* aRow = sf + (size_t)(rowBase + m) * QD;
  v2f aFrag[16];
  #pragma unroll
  for (int ks = 0; ks < 16; ++ks)
    aFrag[ks] = *(const v2f*)(aRow + 4 * ks + 2 * kh);

  float sqRow[8];
  #pragma unroll
  for (int v = 0; v < 8; ++v) sqRow[v] = sq[rowBase + v + 8 * kh];

  // Selection role: 16 threads per row; thread (sr,sub) scans cols sub+16*j.
  const int sr  = tid >> 4;
  const int sub = tid & 15;
  float best[KNN];
  #pragma unroll
  for (int q = 0; q < KNN; ++q) best[q] = 3.4e38f;
  float bmax = 3.4e38f;

  for (int chunk = 0; chunk < NTOT / CHUNK; ++chunk) {
    const int colBase = chunk * CHUNK + wave * 16;
    const float* bRow = sf + (size_t)(colBase + m) * QD;
    const float sqCol = sq[colBase + m];

    v8f c = {0.f, 0.f, 0.f, 0.f, 0.f, 0.f, 0.f, 0.f};
    #pragma unroll
    for (int ks = 0; ks < 16; ++ks) {
      v2f bFrag = *(const v2f*)(bRow + 4 * ks + 2 * kh);
      c = __builtin_amdgcn_wmma_f32_16x16x4_f32(
          /*neg_a=*/false, aFrag[ks], /*neg_b=*/false, bFrag,
          /*c_mod=*/(short)0, c, /*reuse_a=*/false, /*reuse_b=*/false);
    }

    // dist = sqrt(max(|a|^2+|b|^2-2ab, 1e-12)); C/D layout: M=v+8*kh, N=m
    #pragma unroll
    for (int v = 0; v < 8; ++v) {
      float d2 = fmaxf(sqRow[v] + sqCol - 2.0f * c[v], 1e-12f);
      dch[v + 8 * kh][wave * 16 + m] = __builtin_sqrtf(d2);
    }
    __syncthreads();

    // scan 8 candidates for my row; cached-max makes common path 1 compare
    #pragma unroll
    for (int j = 0; j < CHUNK / 16; ++j) {
      float d = dch[sr][sub + 16 * j];
      if (d < bmax) {
        int mi = 0; float mx = best[0];
        #pragma unroll
        for (int q = 1; q < KNN; ++q)
          if (best[q] > mx) { mx = best[q]; mi = q; }
        best[mi] = d;
        mx = best[0];
        #pragma unroll
        for (int q = 1; q < KNN; ++q) mx = fmaxf(mx, best[q]);
        bmax = mx;
      }
    }
    __syncthreads();
  }

  // Merge the 16 per-thread lists of each row -> global top-16 -> mean.
  #pragma unroll
  for (int q = 0; q < KNN; ++q) mergeBuf[sr][sub * KNN + q] = best[q];
  __syncthreads();
  if (tid < 16) {
    float fin[KNN];
    #pragma unroll
    for (int q = 0; q < KNN; ++q) fin[q] = 3.4e38f;
    float fmax_ = 3.4e38f;
    for (int s = 0; s < 16 * KNN; ++s) {
      float d = mergeBuf[tid][s];
      if (d < fmax_) {
        int mi = 0; float mx = fin[0];
        #pragma unroll
        for (int q = 1; q < KNN; ++q)
          if (fin[q] > mx) { mx = fin[q]; mi = q; }
        fin[mi] = d;
        mx = fin[0];
        #pragma unroll
        for (int q = 1; q < KNN; ++q) mx = fmaxf(mx, fin[q]);
        fmax_ = mx;
      }
    }
    float s = 0.0f;
    #pragma unroll
    for (int q = 0; q < KNN; ++q) s += fin[q];
    raw[rowBase + tid] = s * (1.0f / KNN);
  }
}

// ---------------------------------------------------------------------------
// Kernel C: StreamNorm (single block) + final add with reward.
// mag = 0.99 + 0.01 * mean(|raw|); out = reward[:, :512, 0] + raw / (mag+eps)
// ---------------------------------------------------------------------------
__global__ __launch_bounds__(1024) void vib_final_kernel(
    const float* __restrict__ raw, const float* __restrict__ reward,
    float* __restrict__ out) {
  __shared__ float red[1024];
  __shared__ float invMag;
  const int tid = threadIdx.x;
  float s = 0.0f;
  for (int n = tid; n < NTOT; n += 1024) s += fabsf(raw[n]);
  red[tid] = s;
  __syncthreads();
  for (int off = 512; off > 0; off >>= 1) {
    if (tid < off) red[tid] += red[tid + off];
    __syncthreads();
  }
  if (tid == 0) {
    float mag = 0.99f + 0.01f * (red[0] * (1.0f / (float)NTOT));
    invMag = 1.0f / (mag + 1e-8f);
  }
  __syncthreads();
  const float im = invMag;
  for (int n = tid; n < NTOT; n += 1024) {
    const int b = n >> 9, i = n & 511;
    out[n] = reward[b * TFULL + i] + raw[n] * im;
  }
}

// ---------------------------------------------------------------------------
extern "C" void kernel_launch(void* const* d_in, const int* in_sizes, int n_in,
                              void* d_out, int out_size, void* d_ws, size_t ws_size,
                              hipStream_t stream) {
  const float* feat   = (const float*)d_in[0];  // [16,513,1536]
  const float* reward = (const float*)d_in[1];  // [16,513,1]
  const float* proj   = (const float*)d_in[2];  // [1536,64]
  float* out = (float*)d_out;                   // [16,512,1]

  float* sf  = (float*)d_ws;                    // 8192*64 f32 = 2 MB
  float* sq  = sf + (size_t)NTOT * QD;          // 8192 f32
  float* raw = sq + NTOT;                       // 8192 f32

  vib_proj_kernel<<<NTOT / 4, 256, 0, stream>>>(feat, proj, sf, sq);
  vib_knn_kernel <<<NTOT / 16, 256, 0, stream>>>(sf, sq, raw);
  vib_final_kernel<<<1, 1024, 0, stream>>>(raw, reward, out);
}